// CheckboxDetector_12695923327043
// MI455X (gfx1250) — compile-verified
//
#include <hip/hip_runtime.h>
#include <hip/hip_bf16.h>

#define N_BOX 4096
#define THRESHOLD 0.7f
#define IOU_THR 0.5f

typedef float v2f __attribute__((ext_vector_type(2)));
typedef float v8f __attribute__((ext_vector_type(8)));

// pointee type expected by __builtin_amdgcn_global_load_async_to_lds_b128
typedef int b128_t __attribute__((vector_size(16)));
typedef __attribute__((address_space(1))) b128_t* glob_b128_p;
typedef __attribute__((address_space(3))) b128_t* lds_b128_p;

#define HAS_ASYNC_LDS                                                   \
    (__has_builtin(__builtin_amdgcn_global_load_async_to_lds_b128) &&   \
     __has_builtin(__builtin_amdgcn_s_wait_asynccnt))

__device__ __forceinline__ unsigned mono_key(float f) {
    unsigned u = __float_as_uint(f);
    return (u & 0x80000000u) ? ~u : (u | 0x80000000u);
}

// ---------------------------------------------------------------------------
// Kernel 1: score + stable descending bitonic sort (one block, 1024 threads),
// gather sorted boxes/conf, emit valid bitmask.
// ---------------------------------------------------------------------------
__global__ __launch_bounds__(1024)
void nms_sort_kernel(const float* __restrict__ conf,    // [N,2]
                     const float4* __restrict__ boxes,  // [N] (x1,y1,x2,y2)
                     float4* __restrict__ bs,           // sorted boxes out
                     float* __restrict__ cs,            // sorted conf out [N,2]
                     unsigned* __restrict__ validbits)  // 128 words
{
    __shared__ unsigned long long keys[N_BOX];
    const int t = threadIdx.x;

    if (t < 128) validbits[t] = 0u;

    for (int e = t; e < N_BOX; e += 1024) {
        float c0 = conf[2 * e], c1 = conf[2 * e + 1];
        bool valid = (c0 > THRESHOLD) || (c1 > THRESHOLD);
        float score = valid ? fmaxf(c0, c1) : -1.0f;
        keys[e] = ((unsigned long long)mono_key(score) << 32) |
                  (unsigned)(N_BOX - 1 - e);   // stable tie-break: low idx first
    }
    __syncthreads();

    // bitonic sort, descending on 64-bit key
    for (int k = 2; k <= N_BOX; k <<= 1) {
        for (int j = k >> 1; j > 0; j >>= 1) {
            for (int e = t; e < N_BOX; e += 1024) {
                int ixj = e ^ j;
                if (ixj > e) {
                    unsigned long long a = keys[e], b = keys[ixj];
                    bool desc = ((e & k) == 0);
                    if (desc ? (a < b) : (a > b)) { keys[e] = b; keys[ixj] = a; }
                }
            }
            __syncthreads();
        }
    }

    for (int e = t; e < N_BOX; e += 1024) {
        int orig = N_BOX - 1 - (int)(unsigned)(keys[e] & 0xFFFFFFFFull);
        float4 bb = boxes[orig];
        bs[e] = bb;
        float c0 = conf[2 * orig], c1 = conf[2 * orig + 1];
        cs[2 * e] = c0; cs[2 * e + 1] = c1;
        bool valid = (c0 > THRESHOLD) || (c1 > THRESHOLD);
        if (valid) atomicOr(&validbits[e >> 5], 1u << (e & 31));
    }
}

// ---------------------------------------------------------------------------
// Kernel 2: suppression bitmask tiles. Each wave32 owns a 16-row x 32-col
// tile. area[i]+area[j] (the union base) is computed EXACTLY with
// V_WMMA_F32_16X16X4_F32: A row = [area_i, 1, 0, 0], B col = [1, area_j, 0, 0].
// Box tiles are staged global->LDS with the gfx1250 async-to-LDS path
// (ASYNCcnt-tracked, no VGPR round-trip). Ballots assemble 32-bit row words
// straight from the WMMA C/D lane layout. FP contraction is disabled in this
// kernel so union = fl(area_sum - fl(iw*ih)) matches the reference bit-exactly.
// ---------------------------------------------------------------------------
__global__ __launch_bounds__(256)
void nms_mask_kernel(const float4* __restrict__ bs,
                     unsigned* __restrict__ mask)      // [4096][128] words
{
    #pragma clang fp contract(off)

    __shared__ float4 rbox[16];
    __shared__ float4 cbox[256];
    __shared__ float  rarea[16];
    __shared__ float  carea[256];

    const int t  = threadIdx.x;
    const int bx = blockIdx.x;   // row tile: rows [16*bx, 16*bx+16)
    const int by = blockIdx.y;   // col group: cols [256*by, 256*by+256)

#if HAS_ASYNC_LDS
    // async global->LDS: each lane moves one float4 (B128) directly into LDS
    __builtin_amdgcn_global_load_async_to_lds_b128(
        (glob_b128_p)&bs[256 * by + t],
        (lds_b128_p)&cbox[t], 0, 0);
    if (t < 16)
        __builtin_amdgcn_global_load_async_to_lds_b128(
            (glob_b128_p)&bs[16 * bx + t],
            (lds_b128_p)&rbox[t], 0, 0);
    __builtin_amdgcn_s_wait_asynccnt(0);   // this wave's async copies done
#else
    if (t < 16) rbox[t] = bs[16 * bx + t];
    cbox[t] = bs[256 * by + t];
#endif
    __syncthreads();                       // all waves' tiles visible
    if (t < 16) {
        float4 r = rbox[t];
        rarea[t] = (r.z - r.x) * (r.w - r.y);
    }
    {
        float4 c = cbox[t];
        carea[t] = (c.z - c.x) * (c.w - c.y);
    }
    __syncthreads();

    const int wv    = t >> 5;
    const int lane  = t & 31;
    const int cbase = wv * 32;           // wave's 32-col stripe within group
    const bool lo   = (lane < 16);

    // A matrix (16x4 f32): lanes 0-15 hold K0,K1 for row M=lane; lanes 16-31
    // hold K2,K3 (zero).
    v2f A;
    A.x = lo ? rarea[lane & 15] : 0.0f;  // K0 = area_row (or K2 = 0)
    A.y = lo ? 1.0f : 0.0f;              // K1 = 1        (or K3 = 0)

    unsigned ball[2][8];

    #pragma unroll
    for (int ph = 0; ph < 2; ++ph) {
        const int ci = cbase + ph * 16 + (lane & 15);
        // B matrix (4x16 f32): lanes 0-15 hold K0,K1 for col N=lane.
        v2f B;
        B.x = lo ? 1.0f : 0.0f;          // K0 row of B = 1
        B.y = lo ? carea[ci] : 0.0f;     // K1 row of B = area_col
        v8f C = {};
        v8f D = __builtin_amdgcn_wmma_f32_16x16x4_f32(
                    false, A, false, B, (short)0, C, false, false);
        // D[rr]: lanes 0-15 -> row rr, col=lane ; lanes 16-31 -> row rr+8,
        // col=lane-16. Value = area_row + area_col (exact f32).

        const float4 cb = cbox[ci];
        const int gj = by * 256 + ci;
        const int rbase = lo ? 0 : 8;

        #pragma unroll
        for (int rr = 0; rr < 8; ++rr) {
            float4 rb = rbox[rbase + rr];
            float iw = fmaxf(fminf(rb.z, cb.z) - fmaxf(rb.x, cb.x), 0.0f);
            float ih = fmaxf(fminf(rb.w, cb.w) - fmaxf(rb.y, cb.y), 0.0f);
            float inter = iw * ih;            // fl(iw*ih)      (contract off)
            float uni   = D[rr] - inter;      // fl(sum - inter) (contract off)
            float den = fmaxf(uni, 1e-9f);
            int gi = bx * 16 + rbase + rr;
            bool p = ((inter / den) > IOU_THR) && (gj > gi);
            ball[ph][rr] = (unsigned)__ballot(p);
        }
    }

    // Assemble row words. ballot bits[15:0] = row rr (cols ph*16..+15),
    // bits[31:16] = row rr+8 (same cols).
    unsigned myword = 0;
    #pragma unroll
    for (int rr = 0; rr < 8; ++rr) {
        unsigned wlo = (ball[0][rr] & 0xFFFFu) | ((ball[1][rr] & 0xFFFFu) << 16);
        unsigned whi = (ball[0][rr] >> 16)     | (ball[1][rr] & 0xFFFF0000u);
        if (lane == rr)     myword = wlo;
        if (lane == rr + 8) myword = whi;
    }
    if (lane < 16)
        mask[(bx * 16 + lane) * 128 + by * 8 + wv] = myword;
}

// ---------------------------------------------------------------------------
// Kernel 3: sequential greedy scan, single wave32. keep bits live in 4
// uint words per lane (128 words total). Rows are fetched unconditionally
// (pipelined + prefetched); the suppression AND is made branchless with a
// broadcast mask, so only the tiny ALU chain is serial.
// ---------------------------------------------------------------------------
__global__ __launch_bounds__(32)
void nms_scan_kernel(const unsigned* __restrict__ mask,
                     const unsigned* __restrict__ validbits,
                     unsigned* __restrict__ keepbits)
{
    const int lane = threadIdx.x;
    uint4 kv = ((const uint4*)validbits)[lane];
    const uint4* rows = (const uint4*)mask;   // row i at rows[i*32 + lane]

    uint4 nxt = rows[lane];                   // row 0
    for (int i = 0; i < N_BOX; ++i) {
        uint4 cur = nxt;
        if (i + 1 < N_BOX) nxt = rows[(i + 1) * 32 + lane];
        if (i + 16 < N_BOX)
            __builtin_prefetch(&rows[(i + 16) * 32 + lane], 0, 0);

        int w = i >> 5, src = w >> 2, sub = w & 3;
        unsigned kword = (sub == 0) ? kv.x : (sub == 1) ? kv.y
                       : (sub == 2) ? kv.z : kv.w;
        kword = __shfl(kword, src, 32);
        unsigned m = ((kword >> (i & 31)) & 1u) ? 0xFFFFFFFFu : 0u;

        kv.x &= ~(cur.x & m);
        kv.y &= ~(cur.y & m);
        kv.z &= ~(cur.z & m);
        kv.w &= ~(cur.w & m);
    }
    ((uint4*)keepbits)[lane] = kv;
}

// ---------------------------------------------------------------------------
// Kernel 4: finalize -> d_out = [bb (N*4) | cc (N*2) | checked (N) | keep (N)]
// ---------------------------------------------------------------------------
__global__ __launch_bounds__(256)
void nms_finalize_kernel(const float4* __restrict__ bs,
                         const float* __restrict__ cs,
                         const unsigned* __restrict__ keepbits,
                         const int* __restrict__ image_h,
                         const int* __restrict__ image_w,
                         float* __restrict__ out)
{
    int p = blockIdx.x * blockDim.x + threadIdx.x;
    if (p >= N_BOX) return;

    bool keep = (keepbits[p >> 5] >> (p & 31)) & 1u;
    // match Python double arithmetic, then f32 promotion
    double sc = fmax((double)image_w[0] / 1280.0, (double)image_h[0] / 1280.0);
    float scaler = (float)sc;

    float4 b = bs[p];
    float bb0 = keep ? floorf(b.x * scaler) : 0.0f;
    float bb1 = keep ? floorf(b.y * scaler) : 0.0f;
    float bb2 = keep ? floorf(b.z * scaler) : 0.0f;
    float bb3 = keep ? floorf(b.w * scaler) : 0.0f;
    out[p * 4 + 0] = bb0;
    out[p * 4 + 1] = bb1;
    out[p * 4 + 2] = bb2;
    out[p * 4 + 3] = bb3;

    float c0 = cs[2 * p], c1 = cs[2 * p + 1];
    float cc0 = keep ? c0 : 0.0f;
    float cc1 = keep ? c1 : 0.0f;
    out[N_BOX * 4 + p * 2 + 0] = cc0;
    out[N_BOX * 4 + p * 2 + 1] = cc1;

    bool checked = keep && (cc0 > cc1);
    out[N_BOX * 6 + p]         = checked ? 1.0f : 0.0f;
    out[N_BOX * 7 + p]         = keep ? 1.0f : 0.0f;
}

// ---------------------------------------------------------------------------
extern "C" void kernel_launch(void* const* d_in, const int* in_sizes, int n_in,
                              void* d_out, int out_size, void* d_ws, size_t ws_size,
                              hipStream_t stream) {
    const float*  conf    = (const float*)d_in[0];   // [4096,2]
    const float4* boxes   = (const float4*)d_in[1];  // [4096,4]
    const int*    image_h = (const int*)d_in[2];
    const int*    image_w = (const int*)d_in[3];
    float* out = (float*)d_out;

    char* ws = (char*)d_ws;
    unsigned* mask      = (unsigned*)(ws);                 // 4096*128*4 = 2 MB
    float4*   bs        = (float4*)  (ws + 2097152);       // 64 KB
    float*    cs        = (float*)   (ws + 2097152 + 65536);          // 32 KB
    unsigned* validbits = (unsigned*)(ws + 2097152 + 65536 + 32768);  // 512 B
    unsigned* keepbits  = (unsigned*)(ws + 2097152 + 65536 + 32768 + 512);

    nms_sort_kernel<<<1, 1024, 0, stream>>>(conf, boxes, bs, cs, validbits);
    nms_mask_kernel<<<dim3(256, 16), 256, 0, stream>>>(bs, mask);
    nms_scan_kernel<<<1, 32, 0, stream>>>(mask, validbits, keepbits);
    nms_finalize_kernel<<<16, 256, 0, stream>>>(bs, cs, keepbits,
                                                image_h, image_w, out);
}